// WC_57183194579297
// MI455X (gfx1250) — compile-verified
//
#include <hip/hip_runtime.h>
#include <hip/hip_bf16.h>
#include <math.h>

#define SEQ     128
#define EMBED   256
#define HIDDEN  512
#define G3      (3 * HIDDEN)   // 1536
#define NCLS    5

#define NBLK_DIR 32            // persistent blocks per direction
#define UNITS    (HIDDEN / NBLK_DIR)   // 16 hidden units per block
#define SCAN_T   192           // scan block threads: 48 rows * 4 k-splits

typedef float  v2f  __attribute__((ext_vector_type(2)));
typedef float  v8f  __attribute__((ext_vector_type(8)));
typedef unsigned int v4u __attribute__((ext_vector_type(4)));
typedef int    v4i  __attribute__((ext_vector_type(4)));
typedef int    v8i  __attribute__((ext_vector_type(8)));

// ---------------- workspace layout (float offsets) ----------------
#define X_OFF   0                                  // x[128][256]
#define XP_OFF  (X_OFF + SEQ * EMBED)              // xp[2][128][1536]
#define HS_OFF  (XP_OFF + 2 * SEQ * G3)            // hs[2][128][512]
#define HB_OFF  (HS_OFF + 2 * SEQ * HIDDEN)        // h_buf[2][2][512]
#define CTR_OFF (HB_OFF + 2 * 2 * HIDDEN)          // int ctr[2][128]
// total ~2.24 MB

__device__ __forceinline__ float sigmoidf_(float x) {
    return 1.0f / (1.0f + expf(-x));
}

// ---------------- init: zero h0 double-buffers and sync counters ----------------
__global__ void gru_init_kernel(float* wsf) {
    int tid = threadIdx.x;
    float* hb = wsf + HB_OFF;
    for (int i = tid; i < 2 * 2 * HIDDEN; i += 256) hb[i] = 0.0f;
    int* ctr = (int*)(wsf + CTR_OFF);
    if (tid < 2 * SEQ) ctr[tid] = 0;
}

// ---------------- embedding gather: x[s] = embed_table[tokens[s]] ----------------
__global__ void gru_embed_kernel(const int* __restrict__ tokens,
                                 const float* __restrict__ table,
                                 float* __restrict__ wsf) {
    int s = blockIdx.x;                 // 0..127
    int t = tokens[s];
    const float4* src = (const float4*)(table + (size_t)t * EMBED);
    float4* dst = (float4*)(wsf + X_OFF + (size_t)s * EMBED);
    dst[threadIdx.x] = src[threadIdx.x];   // 64 threads * float4 = 256 floats
}

// ---------------- input projection GEMM via V_WMMA_F32_16X16X4_F32 ----------------
// xp[dir][s][row] = x[dir? 127-s : s] . w_ih_dir[row] + b_ih_dir[row]
// One wave per 16x16 output tile.  M = seq (8 tiles), N = 1536 (96 tiles), K = 256.
__global__ void gru_input_proj_wmma(const float* __restrict__ wsf_x,
                                    const float* __restrict__ w_ih_f,
                                    const float* __restrict__ b_ih_f,
                                    const float* __restrict__ w_ih_b,
                                    const float* __restrict__ b_ih_b,
                                    float* __restrict__ xp_out) {
    const int wave  = blockIdx.x * 4 + (threadIdx.x >> 5);  // 4 waves / block
    const int lane  = threadIdx.x & 31;
    const int dir   = wave / (8 * 96);
    const int rem   = wave % (8 * 96);
    const int mt    = rem / 96;
    const int nt    = rem % 96;
    const int sbase = mt * 16;
    const int nbase = nt * 16;

    const int lanelo = lane & 15;
    const int khalf  = (lane >> 4) * 2;      // lanes 16-31 hold K= +2,+3

    const float* W = dir ? w_ih_b : w_ih_f;
    const float* B = dir ? b_ih_b : b_ih_f;

    // A fragment: row M = sbase+lanelo of x (reversed sequence for backward dir)
    const int s     = sbase + lanelo;
    const int xrow  = dir ? (SEQ - 1 - s) : s;
    const float* ap = wsf_x + X_OFF + (size_t)xrow * EMBED + khalf;
    // B fragment: B[k][n] = w_ih[nbase+lanelo][k]
    const float* bp = W + (size_t)(nbase + lanelo) * EMBED + khalf;

    v8f acc = {0.f, 0.f, 0.f, 0.f, 0.f, 0.f, 0.f, 0.f};
    #pragma unroll 8
    for (int k0 = 0; k0 < EMBED; k0 += 4) {
        v2f a = *(const v2f*)(ap + k0);
        v2f b = *(const v2f*)(bp + k0);
        acc = __builtin_amdgcn_wmma_f32_16x16x4_f32(
            /*neg_a=*/false, a, /*neg_b=*/false, b,
            /*c_mod=*/(short)0, acc, /*reuse_a=*/false, /*reuse_b=*/false);
    }

    const float bias = B[nbase + lanelo];
    float* xp = xp_out + XP_OFF + (size_t)dir * SEQ * G3;
    const int mofs = (lane >> 4) * 8;     // C/D layout: VGPR v -> M = v + (hi-half ? 8 : 0)
    #pragma unroll
    for (int v = 0; v < 8; ++v) {
        int srow = sbase + v + mofs;
        xp[(size_t)srow * G3 + nbase + lanelo] = acc[v] + bias;
    }
}

// ---------------- persistent recurrent scan ----------------
// 64 blocks: dir = blockIdx.x / 32, chunk = blockIdx.x % 32 (16 hidden units).
// Each block pins its 3x16x512 slice of w_hh in LDS via one 3-D TDM load, then
// runs all 128 steps with a software grid barrier per step (per direction).
__global__ void __launch_bounds__(SCAN_T)
gru_scan_kernel(const float* __restrict__ w_hh_f,
                const float* __restrict__ b_hh_f,
                const float* __restrict__ w_hh_b,
                const float* __restrict__ b_hh_b,
                float* __restrict__ wsf) {
    __shared__ float w_s[3 * UNITS * HIDDEN];   // 96 KB
    __shared__ float h_s[HIDDEN];
    __shared__ float hp_s[3 * UNITS];

    const int tid = threadIdx.x;
    const int dir = blockIdx.x / NBLK_DIR;
    const int cb  = blockIdx.x % NBLK_DIR;
    const int j0  = cb * UNITS;

    const float* W   = dir ? w_hh_b : w_hh_f;
    const float* bhh = dir ? b_hh_b : b_hh_f;

    // ---- TDM: load 3-D tile {K=512} x {units=16} x {gates=3} into LDS ----
    if (tid < 32) {
        unsigned long long ga = (unsigned long long)(const void*)(W + (size_t)j0 * HIDDEN);
        unsigned int lds = (unsigned int)(unsigned long long)(void*)&w_s[0];
        v4u g0;
        g0[0] = 1u;                                           // count=1, user D#
        g0[1] = lds;                                          // lds_addr
        g0[2] = (unsigned int)(ga & 0xFFFFFFFFu);             // global_addr lo
        g0[3] = (unsigned int)(((ga >> 32) & 0x01FFFFFFu) | 0x80000000u); // hi | type=2
        v8i g1;
        g1[0] = (2 << 16);              // data_size = 4 bytes
        g1[1] = (HIDDEN << 16);         // tensor_dim0[15:0] = 512
        g1[2] = (HIDDEN << 16);         // tensor_dim1[15:0] = 512 (covers tile)
        g1[3] = (HIDDEN << 16);         // tile_dim0 = 512
        g1[4] = UNITS | (3 << 16);      // tile_dim1 = 16, tile_dim2 = 3
        g1[5] = HIDDEN;                 // tensor_dim0_stride = 512 elements
        g1[6] = 0;                      // dim1_stride lo16 (262144 & 0xffff = 0)
        g1[7] = (HIDDEN * HIDDEN) >> 16;// tensor_dim1_stride hi = 4  (512*512)
        v4i g2;
        g2[0] = 3;                      // tensor_dim2 = 3 gates
        g2[1] = 0; g2[2] = 0; g2[3] = 0;
        v4i g3 = {0, 0, 0, 0};
        v8i g4 = {0, 0, 0, 0, 0, 0, 0, 0};   // extra group (clang-23 6-arg form)
        __builtin_amdgcn_tensor_load_to_lds(g0, g1, g2, g3, g4, /*cpol=*/0);
        __builtin_amdgcn_s_wait_tensorcnt(0);
    }
    __syncthreads();

    // per-thread row mapping: 48 rows (3 gates x 16 units), 4 k-splits of 128
    const int row_l = tid >> 2;          // 0..47
    const int ksub  = tid & 3;
    const int gate  = row_l >> 4;        // 0..2
    const int jl    = row_l & 15;
    const float myb = bhh[gate * HIDDEN + j0 + jl];

    float* hb  = wsf + HB_OFF + (size_t)dir * 2 * HIDDEN;
    float* hs  = wsf + HS_OFF + (size_t)dir * SEQ * HIDDEN;
    const float* xpd = wsf + XP_OFF + (size_t)dir * SEQ * G3;
    int* ctr = (int*)(wsf + CTR_OFF) + dir * SEQ;

    for (int t = 0; t < SEQ; ++t) {
        float* cur = hb + (t & 1) * HIDDEN;
        float* nxt = hb + ((t + 1) & 1) * HIDDEN;

        // broadcast h into LDS
        for (int i = tid; i < HIDDEN; i += SCAN_T) h_s[i] = cur[i];
        __syncthreads();

        // partial dot: 128 MACs from LDS weight slice
        const float* wr = &w_s[row_l * HIDDEN + ksub * 128];
        const float* hr = &h_s[ksub * 128];
        float acc = 0.f;
        #pragma unroll 8
        for (int k = 0; k < 128; ++k) acc += wr[k] * hr[k];
        acc += __shfl_xor(acc, 1);
        acc += __shfl_xor(acc, 2);
        if (ksub == 0) hp_s[row_l] = acc + myb;
        __syncthreads();

        // gate math for this block's 16 hidden units
        if (tid < UNITS) {
            const int jg = j0 + tid;
            const float hr_ = hp_s[tid];
            const float hz_ = hp_s[UNITS + tid];
            const float hn_ = hp_s[2 * UNITS + tid];
            const float* xr = xpd + (size_t)t * G3;
            const float xrv = xr[jg];
            const float xzv = xr[HIDDEN + jg];
            const float xnv = xr[2 * HIDDEN + jg];
            const float hold = h_s[jg];
            const float r = sigmoidf_(xrv + hr_);
            const float z = sigmoidf_(xzv + hz_);
            const float n = tanhf(xnv + r * hn_);
            const float hnew = (1.0f - z) * n + z * hold;
            nxt[jg] = hnew;
            hs[(size_t)t * HIDDEN + jg] = hnew;
            __threadfence();
        }
        __syncthreads();

        // software grid barrier across the 32 blocks of this direction
        if (tid == 0) {
            atomicAdd(&ctr[t], 1);
            volatile int* vc = (volatile int*)&ctr[t];
            while (*vc < NBLK_DIR) { __builtin_amdgcn_s_sleep(1); }
        }
        __syncthreads();
        __threadfence();
    }
}

// ---------------- classifier + softmax ----------------
__global__ void gru_cls_kernel(const int* __restrict__ I1p,
                               const float* __restrict__ w_cls,
                               const float* __restrict__ b_cls,
                               const float* __restrict__ wsf,
                               float* __restrict__ out) {
    __shared__ float xi[4 * HIDDEN];
    __shared__ float logit_s[NCLS];
    const int tid = threadIdx.x;
    const int I1 = I1p[0];
    const float* hsf = wsf + HS_OFF;                    // fwd scan outputs
    const float* hsb = wsf + HS_OFF + SEQ * HIDDEN;     // bwd scan outputs

    // x_i1 = [ h_fwd[127], h_fwd[I1], h_bwd[0]=hsb[127], h_bwd[I1]=hsb[127-I1] ]
    for (int i = tid; i < 4 * HIDDEN; i += 192) {
        int seg = i >> 9, off = i & (HIDDEN - 1);
        float v;
        if      (seg == 0) v = hsf[(size_t)(SEQ - 1) * HIDDEN + off];
        else if (seg == 1) v = hsf[(size_t)I1 * HIDDEN + off];
        else if (seg == 2) v = hsb[(size_t)(SEQ - 1) * HIDDEN + off];
        else               v = hsb[(size_t)(SEQ - 1 - I1) * HIDDEN + off];
        xi[i] = v;
    }
    __syncthreads();

    if (tid < NCLS * 32) {
        const int c = tid >> 5, lane = tid & 31;
        float acc = 0.f;
        for (int k = lane; k < 4 * HIDDEN; k += 32)
            acc += xi[k] * w_cls[(size_t)c * 4 * HIDDEN + k];
        #pragma unroll
        for (int m = 16; m >= 1; m >>= 1) acc += __shfl_xor(acc, m);
        if (lane == 0) logit_s[c] = acc + b_cls[c];
    }
    __syncthreads();

    if (tid == 0) {
        float mx = logit_s[0];
        #pragma unroll
        for (int c = 1; c < NCLS; ++c) mx = fmaxf(mx, logit_s[c]);
        float s = 0.f, e[NCLS];
        #pragma unroll
        for (int c = 0; c < NCLS; ++c) { e[c] = expf(logit_s[c] - mx); s += e[c]; }
        #pragma unroll
        for (int c = 0; c < NCLS; ++c) out[c] = e[c] / s;
    }
}

extern "C" void kernel_launch(void* const* d_in, const int* in_sizes, int n_in,
                              void* d_out, int out_size, void* d_ws, size_t ws_size,
                              hipStream_t stream) {
    const int*   tokens = (const int*)  d_in[0];
    const int*   I1     = (const int*)  d_in[1];
    const float* table  = (const float*)d_in[2];
    const float* w_ih_f = (const float*)d_in[3];
    const float* w_hh_f = (const float*)d_in[4];
    const float* b_ih_f = (const float*)d_in[5];
    const float* b_hh_f = (const float*)d_in[6];
    const float* w_ih_b = (const float*)d_in[7];
    const float* w_hh_b = (const float*)d_in[8];
    const float* b_ih_b = (const float*)d_in[9];
    const float* b_hh_b = (const float*)d_in[10];
    const float* w_cls  = (const float*)d_in[11];
    const float* b_cls  = (const float*)d_in[12];
    float* out = (float*)d_out;
    float* wsf = (float*)d_ws;

    gru_init_kernel<<<1, 256, 0, stream>>>(wsf);
    gru_embed_kernel<<<SEQ, 64, 0, stream>>>(tokens, table, wsf);
    // 2 dirs * 8 * 96 tiles = 1536 waves, 4 waves/block
    gru_input_proj_wmma<<<384, 128, 0, stream>>>(wsf, w_ih_f, b_ih_f,
                                                 w_ih_b, b_ih_b, wsf);
    gru_scan_kernel<<<2 * NBLK_DIR, SCAN_T, 0, stream>>>(w_hh_f, b_hh_f,
                                                         w_hh_b, b_hh_b, wsf);
    gru_cls_kernel<<<1, 192, 0, stream>>>(I1, w_cls, b_cls, wsf, out);
}